// model_56246891708529
// MI455X (gfx1250) — compile-verified
//
#include <hip/hip_runtime.h>
#include <hip/hip_bf16.h>
#include <math.h>
#include <stdint.h>

typedef __attribute__((ext_vector_type(2))) float v2f;
typedef __attribute__((ext_vector_type(8))) float v8f;

#define N_NODES   100000
#define N_EDGES   1600000
#define N_GRAPHS  128
#define HIDDEN    128
#define IN_FEAT   4

// ---------------- utility ----------------
__global__ void zero_f32(float* __restrict__ p, int n) {
    int t = blockIdx.x * blockDim.x + threadIdx.x;
    if (t < n) p[t] = 0.0f;
}

__global__ void deg_kernel(const int* __restrict__ col, float* __restrict__ deg, int E) {
    int e = blockIdx.x * blockDim.x + threadIdx.x;
    if (e < E) atomicAdd(&deg[col[e]], 1.0f);
}

__global__ void dinv_kernel(float* __restrict__ d, int n) {
    int t = blockIdx.x * blockDim.x + threadIdx.x;
    if (t < n) {
        float v = d[t];
        d[t] = (v > 0.0f) ? rsqrtf(v) : 0.0f;  // deg>=1 when >0, so max(deg,1)=deg
    }
}

__global__ void norm_kernel(const int* __restrict__ row, const int* __restrict__ col,
                            const float* __restrict__ dinv, float* __restrict__ nrm, int E) {
    int e = blockIdx.x * blockDim.x + threadIdx.x;
    if (e < E) nrm[e] = dinv[row[e]] * dinv[col[e]];
}

// hcat[n, 0:4] = x[n, 0:4]; hcat[n, 4:16] = 0
__global__ void packx_kernel(const float* __restrict__ x, float* __restrict__ hcat, int n) {
    int t = blockIdx.x * blockDim.x + threadIdx.x;
    if (t < n) {
        float4 v = *(const float4*)(x + (size_t)t * 4);
        float* r = hcat + (size_t)t * 16;
        *(float4*)(r + 0)  = v;
        *(float4*)(r + 4)  = make_float4(0.f, 0.f, 0.f, 0.f);
        *(float4*)(r + 8)  = make_float4(0.f, 0.f, 0.f, 0.f);
        *(float4*)(r + 12) = make_float4(0.f, 0.f, 0.f, 0.f);
    }
}

// SpMM on 4-wide features inside hcat (row stride 16)
__global__ void spmm4_kernel(const int* __restrict__ row, const int* __restrict__ col,
                             const float* __restrict__ nrm, float* __restrict__ hcat,
                             int srcOff, int dstOff, int E) {
    int e = blockIdx.x * blockDim.x + threadIdx.x;
    if (e < E) {
        int r = row[e], c = col[e];
        float w = nrm[e];
        float4 v = *(const float4*)(hcat + (size_t)r * 16 + srcOff);
        float* d = hcat + (size_t)c * 16 + dstOff;
        atomicAdd(d + 0, w * v.x);
        atomicAdd(d + 1, w * v.y);
        atomicAdd(d + 2, w * v.z);
        atomicAdd(d + 3, w * v.w);
    }
}

// SpMM on 128-wide features: thread per (edge, 4-float chunk); dst pre-zeroed
__global__ void spmm128_kernel(const int* __restrict__ row, const int* __restrict__ col,
                               const float* __restrict__ nrm,
                               const float* __restrict__ src, float* __restrict__ dst) {
    int t = blockIdx.x * blockDim.x + threadIdx.x;
    int e = t >> 5;
    int p = (t & 31) * 4;
    if (e < N_EDGES) {
        int r = row[e], c = col[e];
        float w = nrm[e];
        float4 v = *(const float4*)(src + (size_t)r * 128 + p);
        float* d = dst + (size_t)c * 128 + p;
        atomicAdd(d + 0, w * v.x);
        atomicAdd(d + 1, w * v.y);
        atomicAdd(d + 2, w * v.z);
        atomicAdd(d + 3, w * v.w);
    }
}

// ---------------- fp32 WMMA GEMM ----------------
// O[N,128] (+)= A[N,K] @ W[K,128].  One wave per 16-row strip, full 128-col output.
// W is staged into LDS in "k-paired" layout via per-lane async-to-LDS scatter:
//   element (k,c) -> LDS dword (k>>1)*256 + c*2 + (k&1)
// so every B fragment (W[k][c], W[k+1][c]) is one aligned ds_load_b64.
// INIT: 0 = acc=0, 1 = acc=O (accumulate), 2 = acc=bias broadcast.  RELU at store.
template <int K, int INIT, int RELU>
__global__ __launch_bounds__(256)
void gemm_wmma_f32(const float* __restrict__ A, const float* __restrict__ W,
                   const float* __restrict__ bias, float* __restrict__ O, int nstrips) {
    __shared__ float shW[K * 128];

    constexpr int EITER = (K * 128) / 256;       // elements per thread
    for (int it = 0; it < EITER; ++it) {
        int idx = it * 256 + (int)threadIdx.x;   // (k, c) = (idx>>7, idx&127)
        int k = idx >> 7, c = idx & 127;
        int dstDword = ((k >> 1) << 8) + (c << 1) + (k & 1);
        unsigned ldso = (unsigned)(uintptr_t)&shW[dstDword];  // low 32 bits = LDS offset
        const float* gp = W + idx;
        asm volatile("global_load_async_to_lds_b32 %0, %1, off"
                     :: "v"(ldso), "v"(gp) : "memory");
    }
    asm volatile("s_wait_asynccnt 0x0" ::: "memory");
    __syncthreads();

    int wave = (blockIdx.x * blockDim.x + threadIdx.x) >> 5;
    if (wave < nstrips) {                        // wave-uniform guard: EXEC all-ones inside
        int lane = threadIdx.x & 31;
        int i = lane & 15;                       // tile column / A-fragment row
        int h = lane >> 4;                       // half select
        int r0 = wave * 16;

        const float* arow = A + (size_t)(r0 + i) * K;
        __builtin_prefetch(arow, 0, 1);          // global_prefetch_b8 on this wave's A rows
        if (K > 64) __builtin_prefetch(arow + 64, 0, 1);

        v8f acc[8];
#pragma unroll
        for (int t = 0; t < 8; ++t) {
            if (INIT == 1) {
#pragma unroll
                for (int v = 0; v < 8; ++v)
                    acc[t][v] = O[(size_t)(r0 + v + 8 * h) * 128 + t * 16 + i];
            } else if (INIT == 2) {
                float b = bias[t * 16 + i];
#pragma unroll
                for (int v = 0; v < 8; ++v) acc[t][v] = b;
            } else {
#pragma unroll
                for (int v = 0; v < 8; ++v) acc[t][v] = 0.0f;
            }
        }

        // A fragment: a[v] = A[r0+i][k0 + 2h + v]
        v2f a;
        a.x = arow[2 * h];
        a.y = arow[2 * h + 1];
        for (int k0 = 0; k0 < K; k0 += 4) {
            v2f a_next;
            if (k0 + 4 < K) {                    // software-pipeline the single A load
                a_next.x = arow[k0 + 4 + 2 * h];
                a_next.y = arow[k0 + 4 + 2 * h + 1];
            }
            // paired layout: pair row (k0+2h) starts at dword (k0+2h)*128; col c at +2c
            const float* wp = &shW[(k0 + 2 * h) * 128 + i * 2];
#pragma unroll
            for (int t = 0; t < 8; ++t) {
                v2f b = *(const v2f*)(wp + t * 32);   // single ds_load_b64, aligned pair
                acc[t] = __builtin_amdgcn_wmma_f32_16x16x4_f32(
                    false, a, false, b, (short)0, acc[t], false, false);
            }
            a = a_next;
        }

#pragma unroll
        for (int t = 0; t < 8; ++t) {
#pragma unroll
            for (int v = 0; v < 8; ++v) {
                float val = acc[t][v];
                if (RELU) val = fmaxf(val, 0.0f);
                O[(size_t)(r0 + v + 8 * h) * 128 + t * 16 + i] = val;
            }
        }
    }
}

// ---------------- pooling + head ----------------
__global__ void pool_kernel(const float* __restrict__ H, const int* __restrict__ batch,
                            float* __restrict__ pooled) {
    int t = blockIdx.x * blockDim.x + threadIdx.x;
    int n = t >> 5;
    int p = (t & 31) * 4;
    if (n < N_NODES) {
        int b = batch[n];
        float4 v = *(const float4*)(H + (size_t)n * 128 + p);
        float* d = pooled + (size_t)b * 128 + p;
        atomicAdd(d + 0, v.x);
        atomicAdd(d + 1, v.y);
        atomicAdd(d + 2, v.z);
        atomicAdd(d + 3, v.w);
    }
}

__global__ void count_kernel(const int* __restrict__ batch, float* __restrict__ cnt) {
    int n = blockIdx.x * blockDim.x + threadIdx.x;
    if (n < N_NODES) atomicAdd(&cnt[batch[n]], 1.0f);
}

__global__ void head_kernel(const float* __restrict__ pooled, const float* __restrict__ cnt,
                            const float* __restrict__ Wlin, const float* __restrict__ blin,
                            float* __restrict__ out) {
    int g = blockIdx.x * blockDim.x + threadIdx.x;
    if (g < N_GRAPHS) {
        float inv = 1.0f / fmaxf(cnt[g], 1.0f);
        float l0 = blin[0], l1 = blin[1];
        const float* p = pooled + (size_t)g * 128;
        for (int j = 0; j < 128; ++j) {
            float m = p[j] * inv;
            l0 = fmaf(m, Wlin[j * 2 + 0], l0);
            l1 = fmaf(m, Wlin[j * 2 + 1], l1);
        }
        float mx = fmaxf(l0, l1);
        float e0 = expf(l0 - mx), e1 = expf(l1 - mx);
        float s = e0 + e1;
        out[g * 2 + 0] = e0 / s;
        out[g * 2 + 1] = e1 / s;
    }
}

// ---------------- driver ----------------
extern "C" void kernel_launch(void* const* d_in, const int* in_sizes, int n_in,
                              void* d_out, int out_size, void* d_ws, size_t ws_size,
                              hipStream_t stream) {
    const float* x    = (const float*)d_in[0];   // [N,4]
    const float* W1   = (const float*)d_in[1];   // [4,4,128] == [16,128]
    const float* b1   = (const float*)d_in[2];   // [128]
    const float* W2   = (const float*)d_in[3];   // [4,128,128]
    const float* b2   = (const float*)d_in[4];   // [128]
    const float* Wlin = (const float*)d_in[5];   // [128,2]
    const float* blin = (const float*)d_in[6];   // [2]
    const int*   ei   = (const int*)d_in[7];     // [2,E]
    const int*   bat  = (const int*)d_in[8];     // [N]
    float* out = (float*)d_out;

    const int* row = ei;
    const int* col = ei + N_EDGES;

    // workspace layout (floats), all 16B-aligned
    float* ws     = (float*)d_ws;
    float* dinv   = ws;                              // N
    float* nrm    = dinv + N_NODES;                  // E
    float* hcat   = nrm + N_EDGES;                   // N*16
    float* H      = hcat + (size_t)N_NODES * 16;     // N*128
    float* P      = H + (size_t)N_NODES * 128;       // N*128
    float* Q      = P + (size_t)N_NODES * 128;       // N*128
    float* O      = Q + (size_t)N_NODES * 128;       // N*128
    float* pooled = O + (size_t)N_NODES * 128;       // 128*128
    float* cnt    = pooled + N_GRAPHS * HIDDEN;      // 128

    const int TB = 256;
    auto blks = [](long long n, int tb) { return (int)((n + tb - 1) / tb); };
    const int nstrips = N_NODES / 16;                // 6250
    const int gemmBlocks = blks((long long)nstrips * 32, TB);

    // --- gcn norm ---
    zero_f32<<<blks(N_NODES, TB), TB, 0, stream>>>(dinv, N_NODES);
    deg_kernel<<<blks(N_EDGES, TB), TB, 0, stream>>>(col, dinv, N_EDGES);
    dinv_kernel<<<blks(N_NODES, TB), TB, 0, stream>>>(dinv, N_NODES);
    norm_kernel<<<blks(N_EDGES, TB), TB, 0, stream>>>(row, col, dinv, nrm, N_EDGES);

    // --- layer 1: pack hops into hcat [N,16], one fused GEMM ---
    packx_kernel<<<blks(N_NODES, TB), TB, 0, stream>>>(x, hcat, N_NODES);
    spmm4_kernel<<<blks(N_EDGES, TB), TB, 0, stream>>>(row, col, nrm, hcat, 0, 4, N_EDGES);
    spmm4_kernel<<<blks(N_EDGES, TB), TB, 0, stream>>>(row, col, nrm, hcat, 4, 8, N_EDGES);
    spmm4_kernel<<<blks(N_EDGES, TB), TB, 0, stream>>>(row, col, nrm, hcat, 8, 12, N_EDGES);
    // H = relu(hcat @ W1flat + b1)
    gemm_wmma_f32<16, 2, 1><<<gemmBlocks, TB, 0, stream>>>(hcat, W1, b1, H, nstrips);

    // --- layer 2 ---
    // O = H @ W2[0] + b2
    gemm_wmma_f32<128, 2, 0><<<gemmBlocks, TB, 0, stream>>>(H, W2, b2, O, nstrips);
    // P = A_norm H ; O += P @ W2[1]
    zero_f32<<<blks((long long)N_NODES * 128, TB), TB, 0, stream>>>(P, N_NODES * 128);
    spmm128_kernel<<<blks((long long)N_EDGES * 32, TB), TB, 0, stream>>>(row, col, nrm, H, P);
    gemm_wmma_f32<128, 1, 0><<<gemmBlocks, TB, 0, stream>>>(P, W2 + 128 * 128, b2, O, nstrips);
    // Q = A_norm P ; O += Q @ W2[2]
    zero_f32<<<blks((long long)N_NODES * 128, TB), TB, 0, stream>>>(Q, N_NODES * 128);
    spmm128_kernel<<<blks((long long)N_EDGES * 32, TB), TB, 0, stream>>>(row, col, nrm, P, Q);
    gemm_wmma_f32<128, 1, 0><<<gemmBlocks, TB, 0, stream>>>(Q, W2 + 2 * 128 * 128, b2, O, nstrips);
    // P = A_norm Q ; O = relu(O + P @ W2[3])
    zero_f32<<<blks((long long)N_NODES * 128, TB), TB, 0, stream>>>(P, N_NODES * 128);
    spmm128_kernel<<<blks((long long)N_EDGES * 32, TB), TB, 0, stream>>>(row, col, nrm, Q, P);
    gemm_wmma_f32<128, 1, 1><<<gemmBlocks, TB, 0, stream>>>(P, W2 + 3 * 128 * 128, b2, O, nstrips);

    // --- mean pool + head ---
    zero_f32<<<blks(N_GRAPHS * HIDDEN + N_GRAPHS, TB), TB, 0, stream>>>(pooled, N_GRAPHS * HIDDEN + N_GRAPHS);
    pool_kernel<<<blks((long long)N_NODES * 32, TB), TB, 0, stream>>>(O, bat, pooled);
    count_kernel<<<blks(N_NODES, TB), TB, 0, stream>>>(bat, cnt);
    head_kernel<<<1, 128, 0, stream>>>(pooled, cnt, Wlin, blin, out);
}